// RRPlus_M11_48515950576137
// MI455X (gfx1250) — compile-verified
//
#include <hip/hip_runtime.h>
#include <hip/hip_bf16.h>

// ---------------------------------------------------------------------------
// Scale-equivariant 1D CNN forward for MI455X (gfx1250, wave32).
// All channel-contraction convs run on V_WMMA_F32_16X16X4_F32 (f32 matrix
// units, reference precision). Lift conv fuses pool4 in-register: WMMA tile t
// covers strided columns 4n+t, so one pool window lives in the four
// accumulators of a single lane (no cross-lane shuffles). Largest
// intermediate is 120MB; later layers fit in the 192MB L2.
// Operand gathers are fully branchless (mask-clamped addresses + v_cndmask on
// values; every spatial length is a power of two). GConvGG keeps an
// LDS-resident K-row decode table (no divisions in the hot loop).
// ---------------------------------------------------------------------------

typedef __attribute__((ext_vector_type(2))) float v2f;
typedef __attribute__((ext_vector_type(8))) float v8f;

#define BN_EPS 2e-5f
#define KDMAX  1840   // max K-dim (layer 9: 204*3*3 = 1836)

// ---------------------------------------------------------------------------
// Lift: out[b,co,s,lp] = max_{q<4} sum_k w1[co,k] * x[b, 4lp+q + (k-39)*2^s] / 2^s
// One wave = 16(co) x 64(full-res l) WMMA tile; tile t handles columns
// n0 + 4n + t so pool4 happens across the 4 accumulators within each lane.
// ---------------------------------------------------------------------------
__global__ void k_lift_wmma(const float* __restrict__ x,   // [8,1,32768]
                            const float* __restrict__ w1,  // [51,79]
                            float* __restrict__ A0)        // [8,51,9,8192]
{
    const int lane = threadIdx.x;           // 0..31
    const int n0   = blockIdx.x * 64;       // full-res column base
    const int m0   = blockIdx.y * 16;       // co base
    const int b    = blockIdx.z / 9;
    const int s    = blockIdx.z % 9;
    const int d    = 1 << s;
    const float invd = 1.0f / (float)d;
    const float* xb = x + (size_t)b * 32768;

    const int  arow  = m0 + (lane & 15);
    const bool rowok = arow < 51;
    const float* wr  = w1 + (size_t)(rowok ? arow : 0) * 79;
    const int  koff  = (lane >> 4) << 1;    // 0 or 2 (K sub-column per lane half)
    const int  lb4   = n0 + ((lane & 15) << 2);  // strided-column base (pool trick)

    v8f acc0 = {}, acc1 = {}, acc2 = {}, acc3 = {};

    for (int k0 = 0; k0 < 80; k0 += 4) {    // K=79 padded to 80
        const int  ka   = k0 + koff;
        const bool k0ok = (ka < 79);
        const bool k1ok = (ka + 1 < 79);
        const float wa0 = wr[k0ok ? ka : 0];        // always-safe load
        const float wa1 = wr[k1ok ? ka + 1 : 0];
        v2f a;
        a.x = (rowok && k0ok) ? wa0 * invd : 0.0f;  // fold 1/2^s into A
        a.y = (rowok && k1ok) ? wa1 * invd : 0.0f;

        const int off0 = (ka - 39) * d;             // SAME pad: left pad = 39*d
        const int off1 = off0 + d;

        v2f b0, b1, b2, b3;
        {   const int i0 = lb4 + off0,     i1 = lb4 + off1;      // tile 0: col 4n+0
            const float v0 = xb[i0 & 32767], v1 = xb[i1 & 32767];
            b0.x = ((unsigned)i0 < 32768u) ? v0 : 0.0f;
            b0.y = ((unsigned)i1 < 32768u) ? v1 : 0.0f; }
        {   const int i0 = lb4 + 1 + off0, i1 = lb4 + 1 + off1;  // tile 1: col 4n+1
            const float v0 = xb[i0 & 32767], v1 = xb[i1 & 32767];
            b1.x = ((unsigned)i0 < 32768u) ? v0 : 0.0f;
            b1.y = ((unsigned)i1 < 32768u) ? v1 : 0.0f; }
        {   const int i0 = lb4 + 2 + off0, i1 = lb4 + 2 + off1;  // tile 2: col 4n+2
            const float v0 = xb[i0 & 32767], v1 = xb[i1 & 32767];
            b2.x = ((unsigned)i0 < 32768u) ? v0 : 0.0f;
            b2.y = ((unsigned)i1 < 32768u) ? v1 : 0.0f; }
        {   const int i0 = lb4 + 3 + off0, i1 = lb4 + 3 + off1;  // tile 3: col 4n+3
            const float v0 = xb[i0 & 32767], v1 = xb[i1 & 32767];
            b3.x = ((unsigned)i0 < 32768u) ? v0 : 0.0f;
            b3.y = ((unsigned)i1 < 32768u) ? v1 : 0.0f; }

        acc0 = __builtin_amdgcn_wmma_f32_16x16x4_f32(false, a, false, b0, (short)0, acc0, false, false);
        acc1 = __builtin_amdgcn_wmma_f32_16x16x4_f32(false, a, false, b1, (short)0, acc1, false, false);
        acc2 = __builtin_amdgcn_wmma_f32_16x16x4_f32(false, a, false, b2, (short)0, acc2, false, false);
        acc3 = __builtin_amdgcn_wmma_f32_16x16x4_f32(false, a, false, b3, (short)0, acc3, false, false);
    }

    // In-lane pool4: acc0..acc3[e] are columns 4p..4p+3 of pooled column
    // p = n0/4 + (lane&15), rows m0 + e + 8*(lane>>4). No shuffles needed.
    const int colp  = (n0 >> 2) + (lane & 15);
    const int rbase = m0 + ((lane >> 4) << 3);
    float* outp = A0 + (((size_t)b * 51 + rbase) * 9 + s) * 8192 + colp;
#pragma unroll
    for (int e = 0; e < 8; ++e) {
        if (rbase + e < 51) {
            const float v = fmaxf(fmaxf(acc0[e], acc1[e]), fmaxf(acc2[e], acc3[e]));
            outp[(size_t)e * (9 * 8192)] = v;   // row stride = 9*8192
        }
    }
}

// ---------------------------------------------------------------------------
// GConvGG as WMMA GEMM: for one (b, s_out):
//   Y[co,l] = sum_{k=(ci,i,t)} W[co,k] * X[ci, s+i, l+(t-1)*2^(s+i)] / 2^(s+i)
// K-row decode (ci,i,t -> base, tap offset, 1/2^sh) precomputed in LDS once
// per block; hot loop is ds broadcast + masked global loads + WMMA.
// Block = 64 threads = 2 waves, each wave owns a 16(co) x 64(l) tile.
// ---------------------------------------------------------------------------
__global__ void __launch_bounds__(64)
k_gg_wmma(const float* __restrict__ X,  // [B,Ci,Hin,L]
          const float* __restrict__ W,  // [Co, Ci*Nk*3]
          float* __restrict__ Y,        // [B,Co,Hout,L]
          int Ci, int Co, int Hin, int Hout, int Nk, int L)
{
    __shared__ int   sBase[KDMAX];   // element offset of row start in Xb
    __shared__ int   sOff[KDMAX];    // (t-1) * 2^sh
    __shared__ float sInv[KDMAX];    // 1 / 2^sh

    const int tid  = threadIdx.x;
    const int lane = tid & 31;
    const int wid  = tid >> 5;
    const int n0   = blockIdx.x * 64;
    const int m0   = blockIdx.y * 32 + wid * 16;
    const int b    = blockIdx.z / Hout;
    const int s    = blockIdx.z % Hout;
    const int nk3  = Nk * 3;
    const int Kd   = Ci * nk3;
    const int Kd4  = (Kd + 3) & ~3;
    const int Lm   = L - 1;                 // L is a power of two
    const float* Xb = X + (size_t)b * Ci * Hin * (size_t)L;

    for (int k = tid; k < Kd; k += 64) {    // one-time decode (divisions here only)
        const int ci = k / nk3;
        const int r  = k - ci * nk3;
        const int i  = r / 3;
        const int t  = r - 3 * i;
        const int sh = s + i;               // input scale row; sh <= Hin-1
        sBase[k] = (ci * Hin + sh) * L;
        sOff[k]  = (t - 1) * (1 << sh);
        sInv[k]  = 1.0f / (float)(1 << sh);
    }
    __syncthreads();

    const int  arow  = m0 + (lane & 15);
    const bool rowok = arow < Co;
    const float* wr  = W + (size_t)(rowok ? arow : 0) * Kd;
    const int  koff  = (lane >> 4) << 1;
    const int  lb    = n0 + (lane & 15);

    v8f acc0 = {}, acc1 = {}, acc2 = {}, acc3 = {};

    for (int k0 = 0; k0 < Kd4; k0 += 4) {
        const int  ka   = k0 + koff;
        const bool k0ok = (ka < Kd);
        const bool k1ok = (ka + 1 < Kd);
        const int  kc0  = k0ok ? ka : 0;
        const int  kc1  = k1ok ? ka + 1 : 0;

        const float wa0 = wr[kc0];
        const float wa1 = wr[kc1];
        v2f a;
        a.x = (rowok && k0ok) ? wa0 : 0.0f;
        a.y = (rowok && k1ok) ? wa1 : 0.0f;

        const int   base0 = sBase[kc0], off0 = sOff[kc0];
        const int   base1 = sBase[kc1], off1 = sOff[kc1];
        const float iv0   = k0ok ? sInv[kc0] : 0.0f;   // zeroes padded K rows
        const float iv1   = k1ok ? sInv[kc1] : 0.0f;

        v2f b0, b1, b2, b3;
        {   const int i0 = lb + off0,      i1 = lb + off1;
            const float v0 = Xb[base0 + (i0 & Lm)], v1 = Xb[base1 + (i1 & Lm)];
            b0.x = ((unsigned)i0 <= (unsigned)Lm) ? v0 * iv0 : 0.0f;
            b0.y = ((unsigned)i1 <= (unsigned)Lm) ? v1 * iv1 : 0.0f; }
        {   const int i0 = lb + 16 + off0, i1 = lb + 16 + off1;
            const float v0 = Xb[base0 + (i0 & Lm)], v1 = Xb[base1 + (i1 & Lm)];
            b1.x = ((unsigned)i0 <= (unsigned)Lm) ? v0 * iv0 : 0.0f;
            b1.y = ((unsigned)i1 <= (unsigned)Lm) ? v1 * iv1 : 0.0f; }
        {   const int i0 = lb + 32 + off0, i1 = lb + 32 + off1;
            const float v0 = Xb[base0 + (i0 & Lm)], v1 = Xb[base1 + (i1 & Lm)];
            b2.x = ((unsigned)i0 <= (unsigned)Lm) ? v0 * iv0 : 0.0f;
            b2.y = ((unsigned)i1 <= (unsigned)Lm) ? v1 * iv1 : 0.0f; }
        {   const int i0 = lb + 48 + off0, i1 = lb + 48 + off1;
            const float v0 = Xb[base0 + (i0 & Lm)], v1 = Xb[base1 + (i1 & Lm)];
            b3.x = ((unsigned)i0 <= (unsigned)Lm) ? v0 * iv0 : 0.0f;
            b3.y = ((unsigned)i1 <= (unsigned)Lm) ? v1 * iv1 : 0.0f; }

        acc0 = __builtin_amdgcn_wmma_f32_16x16x4_f32(false, a, false, b0, (short)0, acc0, false, false);
        acc1 = __builtin_amdgcn_wmma_f32_16x16x4_f32(false, a, false, b1, (short)0, acc1, false, false);
        acc2 = __builtin_amdgcn_wmma_f32_16x16x4_f32(false, a, false, b2, (short)0, acc2, false, false);
        acc3 = __builtin_amdgcn_wmma_f32_16x16x4_f32(false, a, false, b3, (short)0, acc3, false, false);
    }

    const v8f accs[4] = {acc0, acc1, acc2, acc3};
#pragma unroll
    for (int t = 0; t < 4; ++t) {
        const int col = n0 + 16 * t + (lane & 15);
        if (col < L) {
#pragma unroll
            for (int e = 0; e < 8; ++e) {
                const int row = m0 + e + ((lane >> 4) << 3);
                if (row < Co)
                    Y[(((size_t)b * Co + row) * Hout + s) * (size_t)L + col] = accs[t][e];
            }
        }
    }
}

// ---------------------------------------------------------------------------
// BatchNorm (training mode): per-channel sum / sumsq over (B, S, L).
// ---------------------------------------------------------------------------
__global__ void k_zero_stats(float* __restrict__ sum, float* __restrict__ sq)
{
    const int i = threadIdx.x;
    if (i < 512) { sum[i] = 0.0f; sq[i] = 0.0f; }
}

__global__ void k_stats(const float* __restrict__ X, float* __restrict__ sum,
                        float* __restrict__ sq, int C, int SL, int B)
{
    const int c = blockIdx.x;
    float ls = 0.0f, lq = 0.0f;
    for (int b = 0; b < B; ++b) {
        const float* p = X + ((size_t)b * C + c) * (size_t)SL;
        for (int j = blockIdx.y * blockDim.x + threadIdx.x; j < SL;
             j += gridDim.y * blockDim.x) {
            const float v = p[j];
            ls += v; lq += v * v;
        }
    }
    __shared__ float ss[256], qq[256];
    ss[threadIdx.x] = ls; qq[threadIdx.x] = lq;
    __syncthreads();
    for (int w = blockDim.x >> 1; w > 0; w >>= 1) {
        if (threadIdx.x < (unsigned)w) {
            ss[threadIdx.x] += ss[threadIdx.x + w];
            qq[threadIdx.x] += qq[threadIdx.x + w];
        }
        __syncthreads();
    }
    if (threadIdx.x == 0) {
        atomicAdd(&sum[c], ss[0]);
        atomicAdd(&sq[c],  qq[0]);
    }
}

__global__ void k_bn_finalize(const float* __restrict__ sum, const float* __restrict__ sq,
                              const float* __restrict__ g, const float* __restrict__ bb,
                              float* __restrict__ scale, float* __restrict__ shift,
                              int C, float invN)
{
    const int c = blockIdx.x * blockDim.x + threadIdx.x;
    if (c < C) {
        const float m  = sum[c] * invN;
        const float v  = fmaxf(sq[c] * invN - m * m, 0.0f);   // biased var (jnp.var)
        const float sc = g[c] * rsqrtf(v + BN_EPS);
        scale[c] = sc;
        shift[c] = bb[c] - m * sc;
    }
}

// y = max over pool window of relu(scale*x + shift).
// Grid: (ceil(Lout/256), S, B*C) -> index decode is shifts + one scalar %C.
__global__ void k_bn_apply(const float* __restrict__ X, float* __restrict__ Y,
                           const float* __restrict__ scale, const float* __restrict__ shift,
                           int C, int S, int Lin, int pool, int Lout)
{
    const int lo = blockIdx.x * blockDim.x + threadIdx.x;
    if (lo >= Lout) return;
    const int s  = blockIdx.y;
    const int bc = blockIdx.z;               // b*C + c
    const int c  = bc % C;                   // scalar (uniform per block)
    const float sc = scale[c], sh = shift[c];
    const float* xp = X + ((size_t)bc * S + s) * (size_t)Lin + (size_t)lo * pool;
    float m = 0.0f;                          // relu floor
    for (int q = 0; q < pool; ++q) m = fmaxf(m, fmaf(sc, xp[q], sh));
    Y[((size_t)bc * S + s) * (size_t)Lout + lo] = m;
}

// ---------------------------------------------------------------------------
// Global spatial mean (L=32) and the 10-way classifier + (trivial) h-max.
// ---------------------------------------------------------------------------
__global__ void k_mean32(const float* __restrict__ X, float* __restrict__ M)
{
    const int idx = blockIdx.x * blockDim.x + threadIdx.x;   // b*408 + c
    if (idx < 8 * 408) {
        const float* p = X + (size_t)idx * 32;
        float a = 0.0f;
        for (int i = 0; i < 32; ++i) a += p[i];
        M[idx] = a * (1.0f / 32.0f);
    }
}

__global__ void k_classifier(const float* __restrict__ M, const float* __restrict__ w11,
                             float* __restrict__ out)
{
    const int idx = blockIdx.x * blockDim.x + threadIdx.x;   // b*10 + c
    if (idx < 80) {
        const int b = idx / 10, c = idx % 10;
        const float* m = M + b * 408;
        const float* w = w11 + c * 408;
        float acc = 0.0f;
        for (int i = 0; i < 408; ++i) acc = fmaf(w[i], m[i], acc);
        out[idx] = acc;                      // H==1 -> max over h is identity
    }
}

// ---------------------------------------------------------------------------
// Host orchestration
// ---------------------------------------------------------------------------
extern "C" void kernel_launch(void* const* d_in, const int* in_sizes, int n_in,
                              void* d_out, int out_size, void* d_ws, size_t ws_size,
                              hipStream_t stream)
{
    (void)in_sizes; (void)n_in; (void)out_size; (void)ws_size;

    const float* x   = (const float*)d_in[0];
    const float* w1  = (const float*)d_in[1];
    const float* wgt[12];
    for (int i = 2; i <= 11; ++i) wgt[i] = (const float*)d_in[i];   // w2..w11
    const float *g[10], *bb[10];
    for (int i = 0; i < 10; ++i) { g[i] = (const float*)d_in[12 + 2 * i];
                                   bb[i] = (const float*)d_in[13 + 2 * i]; }
    float* out = (float*)d_out;

    // Workspace layout (floats): A0 120MB, 3x32MB ping-pong, stats, mean.
    float* ws  = (float*)d_ws;
    float* A0  = ws;                           // 30,081,024
    float* P   = A0 + 30081024;                // 8,388,608 each
    float* Q   = P + 8388608;
    float* R   = Q + 8388608;
    float* sum = R + 8388608;                  // 512
    float* sq  = sum + 512;                    // 512
    float* sc  = sq + 512;                     // 512
    float* sh  = sc + 512;                     // 512
    float* Mm  = sh + 512;                     // 3264

    auto run_stats = [&](const float* Xp, int C, int S, int L) {
        k_zero_stats<<<dim3(1), dim3(512), 0, stream>>>(sum, sq);
        const int SL = S * L;
        int chunks = (8 * SL + 65535) / 65536;
        if (chunks < 1) chunks = 1; if (chunks > 64) chunks = 64;
        k_stats<<<dim3(C, chunks), dim3(256), 0, stream>>>(Xp, sum, sq, C, SL, 8);
    };
    auto run_fin = [&](int li, int C, int S, int L) {
        const float invN = 1.0f / (float)(8 * S * L);
        k_bn_finalize<<<dim3((C + 63) / 64), dim3(64), 0, stream>>>(
            sum, sq, g[li], bb[li], sc, sh, C, invN);
    };
    auto run_apply = [&](const float* Xp, float* Yp, int C, int S, int Lin, int pool) {
        const int Lout = Lin / pool;
        dim3 grid((Lout + 255) / 256, S, 8 * C);
        k_bn_apply<<<grid, dim3(256), 0, stream>>>(Xp, Yp, sc, sh, C, S, Lin, pool, Lout);
    };
    auto run_gg = [&](const float* Xp, const float* Wp, float* Yp,
                      int Ci, int Co, int Hin, int Hout, int Nk, int L) {
        dim3 grid((L + 63) / 64, (Co + 31) / 32, 8 * Hout);
        k_gg_wmma<<<grid, dim3(64), 0, stream>>>(Xp, Wp, Yp, Ci, Co, Hin, Hout, Nk, L);
    };

    // ---- lift + pool4 (WMMA, fused) -> A0 [8,51,9,8192] ----
    k_lift_wmma<<<dim3(512, 4, 72), dim3(32), 0, stream>>>(x, w1, A0);

    // ---- bn1 + relu + pool4 -> P = A1 [8,51,9,2048] ----
    run_stats(A0, 51, 9, 8192);  run_fin(0, 51, 9, 8192);
    run_apply(A0, P, 51, 9, 8192, 4);

    // ---- layer 2: gg(w2) 9->7, L=2048 ----
    run_gg(P, wgt[2], Q, 51, 51, 9, 7, 3, 2048);
    run_stats(Q, 51, 7, 2048); run_fin(1, 51, 7, 2048); run_apply(Q, R, 51, 7, 2048, 1);

    // ---- layer 3: gg(w3) 7->7, then pool4 -> 512 ----
    run_gg(R, wgt[3], P, 51, 51, 7, 7, 1, 2048);
    run_stats(P, 51, 7, 2048); run_fin(2, 51, 7, 2048); run_apply(P, Q, 51, 7, 2048, 4);

    // ---- layer 4: gg(w4) 7->5, L=512 ----
    run_gg(Q, wgt[4], R, 51, 102, 7, 5, 3, 512);
    run_stats(R, 102, 5, 512); run_fin(3, 102, 5, 512); run_apply(R, P, 102, 5, 512, 1);

    // ---- layer 5: gg(w5) 5->5, then pool4 -> 128 ----
    run_gg(P, wgt[5], Q, 102, 102, 5, 5, 1, 512);
    run_stats(Q, 102, 5, 512); run_fin(4, 102, 5, 512); run_apply(Q, R, 102, 5, 512, 4);

    // ---- layer 6: gg(w6) 5->3, L=128 ----
    run_gg(R, wgt[6], P, 102, 204, 5, 3, 3, 128);
    run_stats(P, 204, 3, 128); run_fin(5, 204, 3, 128); run_apply(P, Q, 204, 3, 128, 1);

    // ---- layer 7: gg(w7) 3->3 ----
    run_gg(Q, wgt[7], R, 204, 204, 3, 3, 1, 128);
    run_stats(R, 204, 3, 128); run_fin(6, 204, 3, 128); run_apply(R, P, 204, 3, 128, 1);

    // ---- layer 8: gg(w8) 3->3, then pool4 -> 32 ----
    run_gg(P, wgt[8], Q, 204, 204, 3, 3, 1, 128);
    run_stats(Q, 204, 3, 128); run_fin(7, 204, 3, 128); run_apply(Q, R, 204, 3, 128, 4);

    // ---- layer 9: gg(w9) 3->1, L=32 ----
    run_gg(R, wgt[9], P, 204, 408, 3, 1, 3, 32);
    run_stats(P, 408, 1, 32); run_fin(8, 408, 1, 32); run_apply(P, Q, 408, 1, 32, 1);

    // ---- layer 10: gg(w10) 1->1 ----
    run_gg(Q, wgt[10], R, 408, 408, 1, 1, 1, 32);
    run_stats(R, 408, 1, 32); run_fin(9, 408, 1, 32); run_apply(R, P, 408, 1, 32, 1);

    // ---- global mean over L, classifier, h-max (trivial) ----
    k_mean32<<<dim3((8 * 408 + 255) / 256), dim3(256), 0, stream>>>(P, Mm);
    k_classifier<<<dim3(1), dim3(128), 0, stream>>>(Mm, wgt[11], out);
}